// QuantumMultiHeadAttention_65481071403866
// MI455X (gfx1250) — compile-verified
//
#include <hip/hip_runtime.h>
#include <math.h>

// ---- problem constants (from reference) ----
#define EDIM 128
#define SLEN 512
#define BATCH 8
#define NHEAD 16
#define DK 8
#define TOKENS (BATCH * SLEN) // 4096

typedef __attribute__((ext_vector_type(2))) float v2f;
typedef __attribute__((ext_vector_type(8))) float v8f;

// D = A(16x4 f32) * B(4x16 f32) + C(16x16 f32), full wave32 op.
__device__ __forceinline__ v8f wmma4(v2f a, v2f b, v8f c) {
    // 8-arg form: (neg_a, A, neg_b, B, c_mod, C, reuse_a, reuse_b)
    return __builtin_amdgcn_wmma_f32_16x16x4_f32(false, a, false, b, (short)0, c,
                                                 false, false);
}

// -----------------------------------------------------------------------------
// Kernel 1: C = x @ W + bias  (16x16 tile per wave), then the closed-form
// quantum head: out[0] = prod_{j=1..7} cos(a_j); out[w>=1] = prod_{j=0..w} cos(a_j)
// where a_j = proj[j] + theta[head][j]. Output layout: dst[(b*H+h)*S*8 + s*8 + w].
// -----------------------------------------------------------------------------
__global__ __launch_bounds__(32)
void qkv_quantum_kernel(const float* __restrict__ x,
                        const float* __restrict__ W,
                        const float* __restrict__ bias,
                        const float* __restrict__ theta,
                        float* __restrict__ dst) {
    const int lane = threadIdx.x;
    const int g    = lane >> 4;   // half-wave group
    const int l16  = lane & 15;
    const int m0   = blockIdx.x * 16;  // token tile
    const int n0   = blockIdx.y * 16;  // output-feature tile

    const float* __restrict__ xrow = x + (m0 + l16) * EDIM; // A row (M = l16)
    const int nA = n0 + l16;                                // B/C column

    // warm L2/WGP$ for the streamed panels (global_prefetch_b8)
    __builtin_prefetch(xrow, 0, 0);
#pragma unroll
    for (int r = 0; r < EDIM; r += 16)
        __builtin_prefetch(&W[r * EDIM + nA], 0, 0);

    v8f c = {0.f, 0.f, 0.f, 0.f, 0.f, 0.f, 0.f, 0.f};
#pragma unroll 8
    for (int k0 = 0; k0 < EDIM; k0 += 4) {
        const int ka = k0 + 2 * g;
        v2f a, b;
        a.x = xrow[ka];
        a.y = xrow[ka + 1];
        b.x = W[ka * EDIM + nA];
        b.y = W[(ka + 1) * EDIM + nA];
        c = wmma4(a, b, c);
    }

    __shared__ float tile[16][17];
    const float bn = bias[nA];
#pragma unroll
    for (int j = 0; j < 8; ++j)
        tile[j + 8 * g][l16] = c[j] + bn;  // C: M = j + 8*g, N = l16
    __syncthreads();

    // each lane handles one (row = l16, head-half = g) pair: 16 rows x 2 heads
    const int head = blockIdx.y * 2 + g;
    float cw[8];
#pragma unroll
    for (int w = 0; w < 8; ++w) {
        const float ang = tile[l16][g * 8 + w] + theta[head * 8 + w];
        cw[w] = __cosf(ang);
    }
    float outv[8];
    float pref = cw[0];
#pragma unroll
    for (int w = 1; w < 8; ++w) { pref *= cw[w]; outv[w] = pref; }
    float suf = cw[7];
#pragma unroll
    for (int w = 6; w >= 1; --w) suf *= cw[w];
    outv[0] = suf;

    const int mg = m0 + l16;
    const int bb = mg >> 9;          // / SLEN
    const int ss = mg & (SLEN - 1);  // % SLEN
    float* o = dst + (((bb * NHEAD + head) * SLEN + ss) * DK);
#pragma unroll
    for (int w = 0; w < 8; ++w) o[w] = outv[w];
}

// -----------------------------------------------------------------------------
// Kernel 2: per (b,h) attention for a 16-query block.
//   scores = qh @ kh^T / sqrt(8)  -> LDS (16 x 512, row stride 516)
//   row softmax: 32 lanes (lane = row + 16*colHalf), float4 LDS passes
//   out = (P @ vh) / rowsum  -> attn_out[(b*S+s)*E + h*8 + w]
// -----------------------------------------------------------------------------
__global__ __launch_bounds__(32)
void attention_kernel(const float* __restrict__ qh,
                      const float* __restrict__ kh,
                      const float* __restrict__ vh,
                      float* __restrict__ out) {
    const int lane = threadIdx.x;
    const int g    = lane >> 4;
    const int l16  = lane & 15;
    const int s0   = blockIdx.x * 16;  // query block
    const int bh   = blockIdx.y;       // 0..127
    const int b    = bh >> 4;
    const int h    = bh & 15;

    const float* __restrict__ qbase = qh + (bh * SLEN + s0) * DK;
    const float* __restrict__ kbase = kh + bh * SLEN * DK;
    const float* __restrict__ vbase = vh + bh * SLEN * DK;

    // warm L2 for the K/V panels of this (b,h)
#pragma unroll 4
    for (int t = 0; t < SLEN; t += 64) {
        __builtin_prefetch(&kbase[(t + l16 * 4) * DK], 0, 0);
        __builtin_prefetch(&vbase[(t + l16 * 4) * DK], 0, 0);
    }

    // persistent A (16 queries x 8 dims): two K-steps
    v2f a0, a1;
    {
        const float* qr = qbase + l16 * DK;
        a0.x = qr[2 * g];     a0.y = qr[2 * g + 1];
        a1.x = qr[4 + 2 * g]; a1.y = qr[4 + 2 * g + 1];
    }

    // stride 516: 16B-aligned rows (b128-friendly); 516 % 64 == 4 keeps a
    // 16-lane float4 column sweep on 64 distinct banks.
    __shared__ __align__(16) float sc[16][516];
    __shared__ float pmax[32];
    __shared__ float psum[32];
    const float scale = 0.35355339059327373f;  // 1/sqrt(8)

    for (int t0 = 0; t0 < SLEN; t0 += 16) {
        const float* kr = kbase + (t0 + l16) * DK;  // B column = key t0+l16
        v2f b0, b1;
        b0.x = kr[2 * g];     b0.y = kr[2 * g + 1];
        b1.x = kr[4 + 2 * g]; b1.y = kr[4 + 2 * g + 1];
        v8f c = {0.f, 0.f, 0.f, 0.f, 0.f, 0.f, 0.f, 0.f};
        c = wmma4(a0, b0, c);
        c = wmma4(a1, b1, c);
#pragma unroll
        for (int j = 0; j < 8; ++j)
            sc[j + 8 * g][t0 + l16] = c[j] * scale;
    }
    __syncthreads();

    // ---- softmax: lane (l16 = row, g = column half) scans 256 cols as float4
    {
        const int c0 = g * 256;
        float m = -3.4e38f;
#pragma unroll 4
        for (int t = 0; t < 256; t += 4) {
            const float4 v = *reinterpret_cast<const float4*>(&sc[l16][c0 + t]);
            m = fmaxf(m, fmaxf(fmaxf(v.x, v.y), fmaxf(v.z, v.w)));
        }
        pmax[lane] = m;
        __syncthreads();
        const float rmax = fmaxf(pmax[l16], pmax[l16 + 16]);
        float sum = 0.f;
#pragma unroll 4
        for (int t = 0; t < 256; t += 4) {
            float4 v = *reinterpret_cast<float4*>(&sc[l16][c0 + t]);
            v.x = __expf(v.x - rmax);
            v.y = __expf(v.y - rmax);
            v.z = __expf(v.z - rmax);
            v.w = __expf(v.w - rmax);
            *reinterpret_cast<float4*>(&sc[l16][c0 + t]) = v;
            sum += (v.x + v.y) + (v.z + v.w);
        }
        psum[lane] = sum;
        __syncthreads();
    }

    // ---- P(16x512) @ vh(512x8): K = keys, N = head dim (cols 8..15 fed zero)
    v8f c = {0.f, 0.f, 0.f, 0.f, 0.f, 0.f, 0.f, 0.f};
    const bool nvalid = (l16 < DK);
    for (int k0 = 0; k0 < SLEN; k0 += 4) {
        const int kk = k0 + 2 * g;
        v2f a, bv;
        a.x = sc[l16][kk];
        a.y = sc[l16][kk + 1];
        bv.x = nvalid ? vbase[kk * DK + l16] : 0.f;
        bv.y = nvalid ? vbase[(kk + 1) * DK + l16] : 0.f;
        c = wmma4(a, bv, c);
    }

    if (nvalid) {
#pragma unroll
        for (int j = 0; j < 8; ++j) {
            const int m = j + 8 * g;
            const float rs = psum[m] + psum[m + 16];
            out[(b * SLEN + s0 + m) * EDIM + h * DK + l16] = c[j] / rs;
        }
    }
}

// -----------------------------------------------------------------------------
// Kernel 3: y = attn_out @ Wo + bo  (straight WMMA GEMM into d_out)
// -----------------------------------------------------------------------------
__global__ __launch_bounds__(32)
void out_proj_kernel(const float* __restrict__ A,
                     const float* __restrict__ Wo,
                     const float* __restrict__ bo,
                     float* __restrict__ y) {
    const int lane = threadIdx.x;
    const int g    = lane >> 4;
    const int l16  = lane & 15;
    const int m0   = blockIdx.x * 16;
    const int n0   = blockIdx.y * 16;

    const float* __restrict__ arow = A + (m0 + l16) * EDIM;
    const int nA = n0 + l16;

    __builtin_prefetch(arow, 0, 0);
#pragma unroll
    for (int r = 0; r < EDIM; r += 16)
        __builtin_prefetch(&Wo[r * EDIM + nA], 0, 0);

    v8f c = {0.f, 0.f, 0.f, 0.f, 0.f, 0.f, 0.f, 0.f};
#pragma unroll 8
    for (int k0 = 0; k0 < EDIM; k0 += 4) {
        const int ka = k0 + 2 * g;
        v2f a, b;
        a.x = arow[ka];
        a.y = arow[ka + 1];
        b.x = Wo[ka * EDIM + nA];
        b.y = Wo[(ka + 1) * EDIM + nA];
        c = wmma4(a, b, c);
    }

    const float bn = bo[nA];
#pragma unroll
    for (int j = 0; j < 8; ++j)
        y[(m0 + j + 8 * g) * EDIM + nA] = c[j] + bn;
}

// -----------------------------------------------------------------------------
extern "C" void kernel_launch(void* const* d_in, const int* in_sizes, int n_in,
                              void* d_out, int out_size, void* d_ws, size_t ws_size,
                              hipStream_t stream) {
    const float* x     = (const float*)d_in[0];
    const float* Wq    = (const float*)d_in[1];
    const float* bq    = (const float*)d_in[2];
    const float* Wk    = (const float*)d_in[3];
    const float* bk    = (const float*)d_in[4];
    const float* Wv    = (const float*)d_in[5];
    const float* bv    = (const float*)d_in[6];
    const float* Wo    = (const float*)d_in[7];
    const float* bo    = (const float*)d_in[8];
    const float* theta = (const float*)d_in[9];

    // workspace: qh | kh | vh | attn_out  (each head-tensor 128*512*8 f32 = 2MB)
    const size_t HEADSZ = (size_t)BATCH * NHEAD * SLEN * DK;  // 524288 floats
    float* qh = (float*)d_ws;
    float* kh = qh + HEADSZ;
    float* vh = kh + HEADSZ;
    float* ao = vh + HEADSZ;  // TOKENS * EDIM floats; total 8 MB of d_ws

    dim3 blk(32, 1, 1);
    dim3 g1(TOKENS / 16, EDIM / 16, 1);  // (256, 8)
    qkv_quantum_kernel<<<g1, blk, 0, stream>>>(x, Wq, bq, theta, qh);
    qkv_quantum_kernel<<<g1, blk, 0, stream>>>(x, Wk, bk, theta, kh);
    qkv_quantum_kernel<<<g1, blk, 0, stream>>>(x, Wv, bv, theta, vh);

    dim3 g2(SLEN / 16, BATCH * NHEAD, 1);  // (32, 128)
    attention_kernel<<<g2, blk, 0, stream>>>(qh, kh, vh, ao);

    dim3 g3(TOKENS / 16, EDIM / 16, 1);
    out_proj_kernel<<<g3, blk, 0, stream>>>(ao, Wo, bo, (float*)d_out);
}